// MIX_jit_56264071578239
// MI455X (gfx1250) — compile-verified
//
#include <hip/hip_runtime.h>
#include <stdint.h>

// Problem constants (fixed by the reference).
#define B_DIM 4096
#define N_DIM 4096
#define NTILE 256     // neurons per block (== threads per block)
#define NTHREADS 256  // 8 wave32 waves per block
#define BPB 64        // batch rows per block

#define LOG2E 1.44269504088896340736f

// CDNA5 async global->LDS path (probe-guarded; falls back to plain copies).
#if defined(__HIP_DEVICE_COMPILE__) && defined(__gfx1250__)
#if __has_builtin(__builtin_amdgcn_global_load_async_to_lds_b128)
#define USE_ASYNC_LDS 1
#endif
#endif

typedef int v4i_t __attribute__((ext_vector_type(4)));
typedef __attribute__((address_space(1))) v4i_t* g_v4i_ptr;
typedef __attribute__((address_space(3))) v4i_t* l_v4i_ptr;

__device__ __forceinline__ float fast_exp2(float x) {
#if __has_builtin(__builtin_amdgcn_exp2f)
  return __builtin_amdgcn_exp2f(x);   // v_exp_f32
#else
  return exp2f(x);
#endif
}

__device__ __forceinline__ float fast_rcp(float x) {
#if __has_builtin(__builtin_amdgcn_rcpf)
  return __builtin_amdgcn_rcpf(x);    // v_rcp_f32
#else
  return 1.0f / x;
#endif
}

__device__ __forceinline__ float fast_sigmoid(float x) {
  // 1/(1+e^-x) = rcp(1 + exp2(-log2(e)*x))
  return fast_rcp(1.0f + fast_exp2(-LOG2E * x));
}

// 16-byte global -> LDS copy; async path on gfx1250.
// Signature (from hipcc diagnostic): (v4i AS1* src, v4i AS3* dst, imm off, imm cpol).
// Generic->AS1 is numerically identity; generic LDS ptr low 32 bits == LDS byte
// offset (ISA 10.2 aperture mapping), so integer round-trips are exact.
__device__ __forceinline__ void cp16_g2l(const float* __restrict__ g, float* l) {
#if defined(USE_ASYNC_LDS)
  __builtin_amdgcn_global_load_async_to_lds_b128(
      (g_v4i_ptr)(uintptr_t)g,
      (l_v4i_ptr)(uint32_t)(uintptr_t)l,
      /*offset=*/0, /*cpol=*/0);
#else
  *(float4*)l = *(const float4*)g;
#endif
}

__global__ void __launch_bounds__(NTHREADS, 2)
moe_gate_kernel(const float* __restrict__ s,
                const float* __restrict__ W1,
                const float* __restrict__ b1,
                const float* __restrict__ W2,
                const float* __restrict__ b2,
                float* __restrict__ out) {
  // 40 KB of per-tile parameters in LDS (WGP has 320 KB -> several blocks resident).
  __shared__ __align__(16) float lw1[NTILE * 16];
  __shared__ __align__(16) float lw2[NTILE * 16];
  __shared__ __align__(16) float lb1[NTILE * 4];
  __shared__ __align__(16) float lb2[NTILE * 4];

  const int tid = threadIdx.x;
  const int n0 = blockIdx.x * NTILE;
  const int n = n0 + tid;
  const int b0 = blockIdx.y * BPB;

  // ---- Stage params for 256 neurons: 10 x b128 async transfers per thread ----
  {
    const float* gW1 = W1 + (size_t)n0 * 16;
    const float* gW2 = W2 + (size_t)n0 * 16;
    const float* gb1 = b1 + (size_t)n0 * 4;
    const float* gb2 = b2 + (size_t)n0 * 4;
#pragma unroll
    for (int i = 0; i < 4; ++i) {
      const int j = tid + NTHREADS * i;           // 1024 x 16B = 16 KB each
      cp16_g2l(gW1 + j * 4, &lw1[j * 4]);
      cp16_g2l(gW2 + j * 4, &lw2[j * 4]);
    }
    cp16_g2l(gb1 + tid * 4, &lb1[tid * 4]);       // 4 KB each
    cp16_g2l(gb2 + tid * 4, &lb2[tid * 4]);
  }

#if defined(USE_ASYNC_LDS)
#if __has_builtin(__builtin_amdgcn_s_wait_asynccnt)
  __builtin_amdgcn_s_wait_asynccnt(0);
#else
  asm volatile("s_wait_asynccnt 0x0" ::: "memory");
#endif
#endif
  __syncthreads();

  // ---- Each thread owns one neuron: pull its 40 params into registers once ----
  float w1r[16], w2r[16], b1r[4], b2r[4];
#pragma unroll
  for (int i = 0; i < 16; ++i) w1r[i] = lw1[tid * 16 + i];
#pragma unroll
  for (int i = 0; i < 16; ++i) w2r[i] = lw2[tid * 16 + i];
#pragma unroll
  for (int i = 0; i < 4; ++i) b1r[i] = lb1[tid * 4 + i];
#pragma unroll
  for (int i = 0; i < 4; ++i) b2r[i] = lb2[tid * 4 + i];

  // ---- Stream 64 batch rows; fully coalesced 4B/lane in and out ----
  // s / out are touched exactly once each -> non-temporal so the 128 MB stream
  // doesn't evict the L2-resident 640 KB parameter set.
#pragma unroll 2
  for (int j = 0; j < BPB; ++j) {
    const int b = b0 + j;
    const size_t idx = (size_t)b * N_DIM + n;
    const float x = __builtin_nontemporal_load(&s[idx]);
    // Speculative prefetch 8 rows ahead (global_prefetch_b8; OOB overshoot at the
    // tail is dropped silently per ISA 10.5 -- prefetch never faults the wave).
    __builtin_prefetch(&s[idx + (size_t)8 * N_DIM], 0, 0);

    // Candidate activations (relu, sigmoid, antirelu, identity)
    const float a0 = fmaxf(x, 0.0f);
    const float a1 = fast_sigmoid(x);
    const float a2 = fminf(x, 0.0f);
    const float a3 = x;

    // Layer 1: h = sigmoid(acts @ W1[n] + b1[n]);  W1 layout [n][k][h]
    float h[4];
#pragma unroll
    for (int hh = 0; hh < 4; ++hh) {
      float t = b1r[hh];
      t = fmaf(a0, w1r[0 * 4 + hh], t);
      t = fmaf(a1, w1r[1 * 4 + hh], t);
      t = fmaf(a2, w1r[2 * 4 + hh], t);
      t = fmaf(a3, w1r[3 * 4 + hh], t);
      h[hh] = fast_sigmoid(t);
    }

    // Layer 2: logits = h @ W2[n] + b2[n];  W2 layout [n][h][k]
    float l[4];
#pragma unroll
    for (int k = 0; k < 4; ++k) {
      float t = b2r[k];
      t = fmaf(h[0], w2r[0 * 4 + k], t);
      t = fmaf(h[1], w2r[1 * 4 + k], t);
      t = fmaf(h[2], w2r[2 * 4 + k], t);
      t = fmaf(h[3], w2r[3 * 4 + k], t);
      l[k] = t;
    }

    // Softmax over 4 logits. h in (0,1), |W2|,|b2| < 0.5 -> |logit| <= 2.5, so
    // exp cannot overflow and the max-subtraction is unnecessary (identical
    // result mathematically, ~10% fewer VALU ops).
    const float e0 = fast_exp2(l[0] * LOG2E);
    const float e1 = fast_exp2(l[1] * LOG2E);
    const float e2 = fast_exp2(l[2] * LOG2E);
    const float e3 = fast_exp2(l[3] * LOG2E);
    const float denom = (e0 + e1) + (e2 + e3);
    float num = e0 * a0;
    num = fmaf(e1, a1, num);
    num = fmaf(e2, a2, num);
    num = fmaf(e3, a3, num);
    __builtin_nontemporal_store(num * fast_rcp(denom), &out[idx]);
  }
}

extern "C" void kernel_launch(void* const* d_in, const int* in_sizes, int n_in,
                              void* d_out, int out_size, void* d_ws, size_t ws_size,
                              hipStream_t stream) {
  const float* s  = (const float*)d_in[0];
  const float* W1 = (const float*)d_in[1];
  const float* b1 = (const float*)d_in[2];
  const float* W2 = (const float*)d_in[3];
  const float* b2 = (const float*)d_in[4];
  float* out = (float*)d_out;

  dim3 grid(N_DIM / NTILE, B_DIM / BPB);  // (16, 64) = 1024 blocks
  dim3 block(NTHREADS);
  moe_gate_kernel<<<grid, block, 0, stream>>>(s, W1, b1, W2, b2, out);
}